// MLP_GCNLayer_19172734009936
// MI455X (gfx1250) — compile-verified
//
#include <hip/hip_runtime.h>

typedef float v2f __attribute__((ext_vector_type(2)));
typedef float v8f __attribute__((ext_vector_type(8)));

#define IN_F  512
#define OUT_F 256

// ---------------------------------------------------------------------------
// Kernel 1: h = feat @ W^T + b  via V_WMMA_F32_16X16X4_F32
// One wave = one 16x16 output tile. 8 waves per block cover 8 consecutive
// n-tiles of the same 16-row m-band (A fragments hit in cache across waves).
// ---------------------------------------------------------------------------
__global__ __launch_bounds__(256) void gemm_wmma_f32(
    const float* __restrict__ feat, const float* __restrict__ W,
    const float* __restrict__ bias, float* __restrict__ h, int nRows)
{
    const int lane        = threadIdx.x & 31;
    const int waveInBlock = threadIdx.x >> 5;          // 0..7
    const int ntile       = ((blockIdx.x & 1) << 3) + waveInBlock;  // 0..15
    const int mtile       = blockIdx.x >> 1;
    const int m0          = mtile * 16;
    if (m0 >= nRows) return;

    const int laneLo  = lane & 15;
    const int kHalf   = (lane >> 4) << 1;              // 0 (lanes 0-15) or 2

    // A fragment (16x4 f32): lane holds row=laneLo, K = k0+kHalf, k0+kHalf+1
    int arow = m0 + laneLo;
    if (arow >= nRows) arow = nRows - 1;               // clamp: keep EXEC all-1s
    // B fragment (4x16 f32): lane holds col=laneLo, same K pair (B[k][n]=W[n][k])
    const int bcol = ntile * 16 + laneLo;

    const float* aPtr = feat + (size_t)arow * IN_F + kHalf;
    const float* bPtr = W    + (size_t)bcol * IN_F + kHalf;

    v8f c = {};
#pragma unroll 8
    for (int k0 = 0; k0 < IN_F; k0 += 4) {
        v2f a = *(const v2f*)(aPtr + k0);
        v2f b = *(const v2f*)(bPtr + k0);
        // (neg_a, A, neg_b, B, c_mod, C, reuse_a, reuse_b)
        c = __builtin_amdgcn_wmma_f32_16x16x4_f32(
                false, a, false, b, (short)0, c, false, false);
    }

    const float bv = bias[bcol];
    // C/D layout: VGPR v: lanes 0-15 -> M=v, N=lane; lanes 16-31 -> M=v+8
    const int rowBase = m0 + ((lane >> 4) << 3);
    float* outBase = h + (size_t)rowBase * OUT_F + bcol;

    if (m0 + 16 <= nRows) {
        // Full tile (always true when nRows % 16 == 0): uniform fast path,
        // one base address + 8 immediate-offset stores (single store clause).
#pragma unroll
        for (int v = 0; v < 8; ++v)
            outBase[(size_t)v * OUT_F] = c[v] + bv;
    } else {
        // Ragged last tile: per-row guard
#pragma unroll
        for (int v = 0; v < 8; ++v) {
            if (rowBase + v < nRows)
                outBase[(size_t)v * OUT_F] = c[v] + bv;
        }
    }
}

// ---------------------------------------------------------------------------
// Kernel 2: zero the output (sum-reduce identity; nodes w/o in-edges get 0)
// ---------------------------------------------------------------------------
__global__ void zero_f32(float* __restrict__ p, size_t n)
{
    size_t i      = (size_t)blockIdx.x * blockDim.x + threadIdx.x;
    size_t stride = (size_t)gridDim.x * blockDim.x;
    for (; i < n; i += stride) p[i] = 0.0f;
}

// ---------------------------------------------------------------------------
// Kernel 3: edge scatter-add. 64 threads per edge, float4 gather of h[src],
// 4 agent-scope relaxed f32 atomics into out[dst]. Output (102 MB) is
// L2-resident on MI455X (192 MB L2), so atomics stay on-chip.
// ---------------------------------------------------------------------------
__global__ __launch_bounds__(256) void scatter_sum(
    const float* __restrict__ h, const int* __restrict__ src,
    const int* __restrict__ dst, float* __restrict__ out, int nEdges)
{
    const long long gtid = (long long)blockIdx.x * blockDim.x + threadIdx.x;
    const int e = (int)(gtid >> 6);          // 64 threads per edge
    if (e >= nEdges) return;
    const int chunk = (int)(gtid & 63);      // 4 floats each -> 256 feats

    const int s = src[e];
    const int d = dst[e];

    const float4 v = *(const float4*)(h + (size_t)s * OUT_F + (chunk << 2));
    float* o = out + (size_t)d * OUT_F + (chunk << 2);

    __hip_atomic_fetch_add(o + 0, v.x, __ATOMIC_RELAXED, __HIP_MEMORY_SCOPE_AGENT);
    __hip_atomic_fetch_add(o + 1, v.y, __ATOMIC_RELAXED, __HIP_MEMORY_SCOPE_AGENT);
    __hip_atomic_fetch_add(o + 2, v.z, __ATOMIC_RELAXED, __HIP_MEMORY_SCOPE_AGENT);
    __hip_atomic_fetch_add(o + 3, v.w, __ATOMIC_RELAXED, __HIP_MEMORY_SCOPE_AGENT);
}

// ---------------------------------------------------------------------------
extern "C" void kernel_launch(void* const* d_in, const int* in_sizes, int n_in,
                              void* d_out, int out_size, void* d_ws, size_t ws_size,
                              hipStream_t stream)
{
    const float* feat = (const float*)d_in[0];   // [N, 512]
    const float* W    = (const float*)d_in[1];   // [256, 512]
    const float* bias = (const float*)d_in[2];   // [256]
    const int*   src  = (const int*)d_in[3];     // [E]
    const int*   dst  = (const int*)d_in[4];     // [E]
    float*       out  = (float*)d_out;           // [N, 256]
    float*       h    = (float*)d_ws;            // scratch [N, 256] f32

    const int nNodes = in_sizes[0] / IN_F;
    const int nEdges = in_sizes[3];

    // GEMM: 2 blocks (of 8 waves) per 16-row m-band cover all 16 n-tiles
    const int mtiles = (nNodes + 15) / 16;
    gemm_wmma_f32<<<dim3(mtiles * 2), 256, 0, stream>>>(feat, W, bias, h, nNodes);

    // Zero output
    zero_f32<<<2048, 256, 0, stream>>>(out, (size_t)out_size);

    // Scatter-add over edges
    const long long totalThreads = (long long)nEdges * 64;
    const int blocks = (int)((totalThreads + 255) / 256);
    scatter_sum<<<blocks, 256, 0, stream>>>(h, src, dst, out, nEdges);
}